// MusicTheoryLoss_33852932227745
// MI455X (gfx1250) — compile-verified
//
#include <hip/hip_runtime.h>

// ---------------------------------------------------------------------------
// MusicTheoryLoss for MI455X (gfx1250, wave32)
//   B=64, T=2048, V=128  -> 131072 rows, 16.78M elements/tensor, fp32.
//   HBM-bound: ~201 MB @ 23.3 TB/s => ~8.6us floor.
//   - float4 nontemporal streaming loads (global_load_b128, NT), unroll 2
//   - one wave32 per row; argmax via packed monotonic keys + shfl_xor tree
//   - QUALITY LUT as packed nibbles in a 64-bit immediate (no memory, no
//     s_wait_loadcnt 0 stall in the hot loop; products n*0.1f are bit-exact
//     vs the reference fp32 table)
//   - wave-level sum folded through V_WMMA_F32_16X16X4_F32 (B = ones row-sum)
//   - deterministic block partials in d_ws + fixed-order final reduce kernel
// ---------------------------------------------------------------------------

typedef __attribute__((ext_vector_type(4))) float v4f;
typedef __attribute__((ext_vector_type(2))) float v2f;
typedef __attribute__((ext_vector_type(8))) float v8f;

#define B_DIM 64
#define T_DIM 2048
#define V_DIM 128
#define NROWS (B_DIM * T_DIM)          // 131072
#define NELEM ((long long)NROWS * V_DIM)
#define TPB 256
#define WAVES_PER_BLOCK (TPB / 32)     // 8
#define NBLOCKS 1024
#define TOTAL_WAVES (NBLOCKS * WAVES_PER_BLOCK)  // 8192
#define ROWS_PER_WAVE (NROWS / TOTAL_WAVES)      // 16

// QUALITY = {1.0,0.5,0.3,0.2,0.1,0.1,0.7,0.0,0.2,0.3,0.4,0.5}
// nibble i = QUALITY[i]*10 :  A,5,3,2,1,1,7,0,2,3,4,5  (idx 0 = lowest nibble)
#define QUALITY_NIBBLES 0x54320711235AULL

// Monotonic mapping: float total order -> unsigned total order
__device__ __forceinline__ unsigned monokey(float v) {
    unsigned u = __float_as_uint(v);
    return (u & 0x80000000u) ? ~u : (u | 0x80000000u);
}

__global__ __launch_bounds__(TPB)
void mtl_main(const float* __restrict__ outputs,
              const float* __restrict__ targets,
              const float* __restrict__ melodies,
              float* __restrict__ partials /* [2*NBLOCKS] */) {
    const int lane = threadIdx.x & 31;
    const int wib  = threadIdx.x >> 5;
    const int gwave = blockIdx.x * WAVES_PER_BLOCK + wib;

    float mse_acc = 0.0f;   // per-lane partial of sum (o-t)^2
    float q_acc   = 0.0f;   // per-wave (lane 0 masked) partial of sum quality

    #pragma unroll 2
    for (int i = 0; i < ROWS_PER_WAVE; ++i) {
        const long long row  = (long long)gwave * ROWS_PER_WAVE + i;
        const long long base = row * V_DIM + (long long)lane * 4;

        // 128-bit non-temporal streaming loads (data is read exactly once)
        v4f o = __builtin_nontemporal_load((const v4f*)(outputs  + base));
        v4f t = __builtin_nontemporal_load((const v4f*)(targets  + base));
        v4f m = __builtin_nontemporal_load((const v4f*)(melodies + base));

        v4f d = o - t;
        mse_acc += d.x * d.x + d.y * d.y + d.z * d.z + d.w * d.w;

        // --- per-row argmax over V=128 (4 elems/lane, 32 lanes) ---
        // key = (monotonic(value) << 32) | (127 - idx):
        //   max key == max value, ties -> lowest index (JAX argmax semantics)
        unsigned long long hk = 0ull, mk = 0ull;
        #pragma unroll
        for (int j = 0; j < 4; ++j) {
            const unsigned idx = (unsigned)(lane * 4 + j);
            unsigned long long k1 =
                ((unsigned long long)monokey(o[j]) << 32) | (unsigned long long)(127u - idx);
            unsigned long long k2 =
                ((unsigned long long)monokey(m[j]) << 32) | (unsigned long long)(127u - idx);
            hk = (k1 > hk) ? k1 : hk;
            mk = (k2 > mk) ? k2 : mk;
        }
        #pragma unroll
        for (int off = 16; off > 0; off >>= 1) {
            unsigned long long ho = __shfl_xor(hk, off, 32);
            unsigned long long mo = __shfl_xor(mk, off, 32);
            hk = (ho > hk) ? ho : hk;
            mk = (mo > mk) ? mo : mk;
        }

        // Wave-uniform after reduction: all lanes compute, lane 0 accumulates
        // (select, no EXEC divergence, no memory LUT).
        {
            int h  = 127 - (int)(hk & 0xFFFFFFFFull);
            int mn = 127 - (int)(mk & 0xFFFFFFFFull);
            int diff = mn - h;
            if (diff < 0) diff = -diff;
            int interval = diff % 12;
            float q = (float)((unsigned)(QUALITY_NIBBLES >> (4 * interval)) & 0xFu) * 0.1f;
            q_acc += (lane == 0) ? q : 0.0f;
        }
    }

    // --- wave reduction via V_WMMA_F32_16X16X4_F32 (exact fp32) ---
    // A (16x4 f32) = 2 VGPRs x 32 lanes; accumulator in v0, zero in v1.
    // B = all-ones => D[m][n] = sum_k A[m][k] = acc(lane m) + acc(lane m+16),
    // replicated across all columns n (layout-independent row-sum).
    v2f a_m; a_m.x = mse_acc; a_m.y = 0.0f;
    v2f a_q; a_q.x = q_acc;   a_q.y = 0.0f;
    v2f bones; bones.x = 1.0f; bones.y = 1.0f;
    v8f cz = {};
    v8f dm = __builtin_amdgcn_wmma_f32_16x16x4_f32(
        false, a_m, false, bones, (short)0, cz, false, false);
    v8f dq = __builtin_amdgcn_wmma_f32_16x16x4_f32(
        false, a_q, false, bones, (short)0, cz, false, false);

    // sum D VGPRs: lanes 0-15 hold rows 0..7, lanes 16-31 hold rows 8..15
    float sm = dm[0] + dm[1] + dm[2] + dm[3] + dm[4] + dm[5] + dm[6] + dm[7];
    float sq = dq[0] + dq[1] + dq[2] + dq[3] + dq[4] + dq[5] + dq[6] + dq[7];
    sm += __shfl_xor(sm, 16, 32);   // full wave total on every lane
    sq += __shfl_xor(sq, 16, 32);

    // --- block reduction (deterministic, fixed order) ---
    __shared__ float s_m[WAVES_PER_BLOCK];
    __shared__ float s_q[WAVES_PER_BLOCK];
    if (lane == 0) { s_m[wib] = sm; s_q[wib] = sq; }
    __syncthreads();
    if (threadIdx.x == 0) {
        float bm = 0.0f, bq = 0.0f;
        #pragma unroll
        for (int w = 0; w < WAVES_PER_BLOCK; ++w) { bm += s_m[w]; bq += s_q[w]; }
        partials[blockIdx.x]           = bm;
        partials[NBLOCKS + blockIdx.x] = bq;
    }
}

__global__ __launch_bounds__(TPB)
void mtl_final(const float* __restrict__ partials, float* __restrict__ out) {
    __shared__ double sm[TPB];
    __shared__ double sq[TPB];
    double am = 0.0, aq = 0.0;
    for (int i = threadIdx.x; i < NBLOCKS; i += TPB) {
        am += (double)partials[i];
        aq += (double)partials[NBLOCKS + i];
    }
    sm[threadIdx.x] = am;
    sq[threadIdx.x] = aq;
    __syncthreads();
    for (int s = TPB / 2; s > 0; s >>= 1) {
        if (threadIdx.x < s) {
            sm[threadIdx.x] += sm[threadIdx.x + s];
            sq[threadIdx.x] += sq[threadIdx.x + s];
        }
        __syncthreads();
    }
    if (threadIdx.x == 0) {
        const double mse     = sm[0] / (double)NELEM;   // mean over B*T*V
        const double harmony = sq[0] / (double)NROWS;   // mean over B*T
        out[0] = (float)(1.0 * mse + 1.0 * harmony);    // ALPHA = BETA = 1
    }
}

extern "C" void kernel_launch(void* const* d_in, const int* in_sizes, int n_in,
                              void* d_out, int out_size, void* d_ws, size_t ws_size,
                              hipStream_t stream) {
    (void)in_sizes; (void)n_in; (void)out_size; (void)ws_size;
    const float* outputs  = (const float*)d_in[0];
    const float* targets  = (const float*)d_in[1];
    const float* melodies = (const float*)d_in[2];
    float* partials = (float*)d_ws;     // 2 * NBLOCKS floats = 8 KB
    float* out      = (float*)d_out;

    mtl_main<<<NBLOCKS, TPB, 0, stream>>>(outputs, targets, melodies, partials);
    mtl_final<<<1, TPB, 0, stream>>>(partials, out);
}